// PointNet2MSG_63256278335591
// MI455X (gfx1250) — compile-verified
//
#include <hip/hip_runtime.h>
#include <hip/hip_bf16.h>

// ---------------------------------------------------------------------------
// PointNet++ MSG forward for MI455X (gfx1250), wave32.
// All GEMMs use V_WMMA_F32_16X16X4_F32 (full f32 precision; workload is
// bandwidth/latency bound, so f32 WMMA is the right precision/perf point).
// Deterministic: no float atomics anywhere; BN stats use fixed-order
// per-channel tree reductions.
// ---------------------------------------------------------------------------

typedef float v2f __attribute__((ext_vector_type(2)));
typedef float v8f __attribute__((ext_vector_type(8)));

#define EPSBN 1e-5f

static inline int ceil_div(int a, int b) { return (a + b - 1) / b; }

// ---------------------------------------------------------------------------
// Farthest point sampling: one block per batch. dist[] in LDS, argmax via
// LDS tree with first-index tie-break (matches jnp.argmax semantics).
// ---------------------------------------------------------------------------
__global__ void __launch_bounds__(256) fps_kernel(const float* __restrict__ coords,
                                                  int* __restrict__ out,
                                                  int B, int N, int K)
{
    __shared__ float dist[4096];
    __shared__ float rv[256];
    __shared__ int   ri[256];
    int b = blockIdx.x;
    int tid = threadIdx.x;
    const float* cb = coords + (size_t)b * 3 * N;
    for (int n = tid; n < N; n += 256) dist[n] = 1e10f;
    __syncthreads();
    int cur = 0;
    for (int i = 0; i < K; ++i) {
        if (tid == 0) out[b * K + i] = cur;
        float lx = cb[cur], ly = cb[N + cur], lz = cb[2 * N + cur];
        float best = -1.0f; int bi = 0;
        for (int n = tid; n < N; n += 256) {
            float dx = cb[n] - lx, dy = cb[N + n] - ly, dz = cb[2 * N + n] - lz;
            float d = dx * dx + dy * dy + dz * dz;
            float nd = fminf(dist[n], d);
            dist[n] = nd;
            if (nd > best) { best = nd; bi = n; }
        }
        rv[tid] = best; ri[tid] = bi;
        __syncthreads();
        for (int s = 128; s > 0; s >>= 1) {
            if (tid < s) {
                float v2 = rv[tid + s]; int i2 = ri[tid + s];
                if (v2 > rv[tid] || (v2 == rv[tid] && i2 < ri[tid])) {
                    rv[tid] = v2; ri[tid] = i2;
                }
            }
            __syncthreads();
        }
        cur = ri[0];
        __syncthreads();
    }
}

// ---------------------------------------------------------------------------
// Gather coords by index: dst[b][c][k] = src[b][c][idx[b][k]]
// ---------------------------------------------------------------------------
__global__ void gather_coords_kernel(const float* __restrict__ src,
                                     const int* __restrict__ idx,
                                     float* __restrict__ dst,
                                     int B, int N, int K)
{
    int i = blockIdx.x * blockDim.x + threadIdx.x;
    if (i >= B * 3 * K) return;
    int k = i % K;
    int c = (i / K) % 3;
    int b = i / (3 * K);
    dst[((size_t)b * 3 + c) * K + k] = src[((size_t)b * 3 + c) * N + idx[b * K + k]];
}

// ---------------------------------------------------------------------------
// Ball query + group gather. One wave per (batch, centroid) group.
// Ordered first-S in-radius selection via ballot + prefix popcount, pad with
// first hit (centroid is always in-radius of itself). Writes relative coords
// and optional features into channel-major X[(3+Cf)][B*K*S].
// Neighbor indices cached in registers for the (large-Cf) feature gather.
// ---------------------------------------------------------------------------
#define SMAX 128
__global__ void __launch_bounds__(128) group_kernel(const float* __restrict__ coords,
                                                    const float* __restrict__ feats,
                                                    const int* __restrict__ cidx,
                                                    float* __restrict__ X,
                                                    int B, int N, int K, int S, int Cf,
                                                    float r2)
{
    __shared__ int nbr[4][SMAX];
    int lane = threadIdx.x & 31;
    int wv = threadIdx.x >> 5;
    int gid = blockIdx.x * 4 + wv;
    if (gid >= B * K) return;
    int b = gid / K, k = gid % K;
    int ci = cidx[b * K + k];
    const float* cb = coords + (size_t)b * 3 * N;
    float cx = cb[ci], cy = cb[N + ci], cz = cb[2 * N + ci];

    int count = 0;
    for (int n0 = 0; n0 < N && count < S; n0 += 32) {
        int n = n0 + lane;
        bool inr = false;
        if (n < N) {
            float dx = cb[n] - cx, dy = cb[N + n] - cy, dz = cb[2 * N + n] - cz;
            inr = (dx * dx + dy * dy + dz * dz) < r2;
        }
        unsigned long long m = __ballot(inr);
        int pos = count + __popcll(m & ((1ull << lane) - 1ull));
        if (inr && pos < S) nbr[wv][pos] = n;
        int tot = __popcll(m);
        count = count + tot; if (count > S) count = S;
    }
    __builtin_amdgcn_wave_barrier();
    int first = nbr[wv][0];
    for (int s = count + lane; s < S; s += 32) nbr[wv][s] = first;
    __builtin_amdgcn_wave_barrier();

    // cache this lane's neighbor indices (<= 4 since SMAX = 128)
    int nreg[4];
#pragma unroll
    for (int j = 0; j < 4; ++j) {
        int s = lane + 32 * j;
        nreg[j] = (s < S) ? nbr[wv][s] : 0;
    }

    size_t Mt = (size_t)B * K * S;
    size_t mb = (size_t)gid * S;
#pragma unroll
    for (int j = 0; j < 4; ++j) {
        int s = lane + 32 * j;
        if (s < S) {
            int n = nreg[j];
            X[0 * Mt + mb + s] = cb[n] - cx;
            X[1 * Mt + mb + s] = cb[N + n] - cy;
            X[2 * Mt + mb + s] = cb[2 * N + n] - cz;
        }
    }
    if (feats) {
        for (int c = 0; c < Cf; ++c) {
            const float* fr = feats + (size_t)c * B * N + (size_t)b * N;
            size_t orow = (size_t)(3 + c) * Mt + mb;
#pragma unroll
            for (int j = 0; j < 4; ++j) {
                int s = lane + 32 * j;
                if (s < S) X[orow + s] = fr[nreg[j]];
            }
        }
    }
}

// ---------------------------------------------------------------------------
// WMMA f32 GEMM: Y[O][M] = W[O][C] * X[C][M] + bias[O]
// One 16x16 output tile per wave via V_WMMA_F32_16X16X4_F32 (K stepped by 4).
// A (16x4): lane l: row = l%16, k = j + 2*(l/16), j in {0,1}
// B (4x16): lane l: col = l%16, k = j + 2*(l/16)
// C/D:      vgpr r, lane l: row = r + 8*(l/16), col = l%16
//
// Mapping: all 8 waves of a block share the SAME 16-column m-tile and cover
// 8 consecutive 16-row O-tiles -> their X loads are identical and collapse
// in the WGP cache (X is the dominant HBM traffic; W is small/L2-resident).
// Hot loop has no guards: out-of-range A rows are address-clamped only (they
// feed never-stored D rows). Only the final partial K-step zeroes B (k >= C).
// ---------------------------------------------------------------------------
__global__ void __launch_bounds__(256) gemm_wmma_f32(const float* __restrict__ W,
                                                     const float* __restrict__ X,
                                                     const float* __restrict__ bias,
                                                     float* __restrict__ Y,
                                                     int O, int C, int M)
{
    int lane = threadIdx.x & 31;
    int wv = threadIdx.x >> 5;
    int mBase = blockIdx.x * 16;
    int oBase = (blockIdx.y * 8 + wv) * 16;
    if (mBase >= M) return;        // wave-uniform
    if (oBase >= O) return;        // wave-uniform: EXEC stays all-ones for WMMA
    int half = lane >> 4;
    int l16 = lane & 15;
    size_t strideM = (size_t)M;

    // accumulator init with bias (clamped address; junk rows never stored)
    v8f acc;
#pragma unroll
    for (int r = 0; r < 8; ++r) {
        int o = oBase + r + 8 * half;
        acc[r] = bias[o < O ? o : 0];
    }

    int aRow = oBase + l16;
    int aRowS = aRow < O ? aRow : 0;              // clamp only; no value select
    const float* wptr = W + (size_t)aRowS * C + 2 * half;
    const float* xptr0 = X + (size_t)(2 * half) * strideM + mBase + l16;
    const float* xptr1 = xptr0 + strideM;

    int nFull = C >> 2;
    v2f a, bfr;
    if (nFull > 0) {
        a[0] = wptr[0]; a[1] = wptr[1];
        bfr[0] = xptr0[0]; bfr[1] = xptr1[0];
    }
    for (int i = 0; i < nFull - 1; ++i) {
        v2f an, bn;
        an[0] = wptr[4]; an[1] = wptr[5];
        bn[0] = xptr0[4 * strideM]; bn[1] = xptr1[4 * strideM];
        wptr += 4; xptr0 += 4 * strideM; xptr1 += 4 * strideM;
        acc = __builtin_amdgcn_wmma_f32_16x16x4_f32(false, a, false, bfr,
                                                    (short)0, acc, false, false);
        a = an; bfr = bn;
    }
    if (nFull > 0) {
        acc = __builtin_amdgcn_wmma_f32_16x16x4_f32(false, a, false, bfr,
                                                    (short)0, acc, false, false);
    }

    // remainder (C % 4 != 0): zero B where k >= C (clamped addresses)
    int c0 = nFull << 2;
    if (c0 < C) {
        int k0 = c0 + 2 * half;
        int k1 = k0 + 1;
        int k0s = k0 < C ? k0 : 0;
        int k1s = k1 < C ? k1 : 0;
        v2f ar, br;
        const float* wr = W + (size_t)aRowS * C;
        ar[0] = wr[k0s]; ar[1] = wr[k1s];
        float x0 = X[(size_t)k0s * strideM + mBase + l16];
        float x1 = X[(size_t)k1s * strideM + mBase + l16];
        br[0] = (k0 < C) ? x0 : 0.0f;
        br[1] = (k1 < C) ? x1 : 0.0f;
        acc = __builtin_amdgcn_wmma_f32_16x16x4_f32(false, ar, false, br,
                                                    (short)0, acc, false, false);
    }

#pragma unroll
    for (int r = 0; r < 8; ++r) {
        int o = oBase + r + 8 * half;
        if (o < O) Y[(size_t)o * strideM + (mBase + l16)] = acc[r];
    }
}

// ---------------------------------------------------------------------------
// Per-channel BN stats (training mode, biased var): one block per channel.
// Fixed-order strided accumulation + LDS tree -> deterministic across replays.
// ---------------------------------------------------------------------------
__global__ void __launch_bounds__(256) stats_kernel(const float* __restrict__ Y,
                                                    int M,
                                                    float* __restrict__ mean,
                                                    float* __restrict__ inv_std)
{
    __shared__ float s1[256];
    __shared__ float s2[256];
    int o = blockIdx.x;
    const float* row = Y + (size_t)o * M;
    float s = 0.0f, ss = 0.0f;
    for (int i = threadIdx.x; i < M; i += 256) {
        float v = row[i];
        s += v; ss += v * v;
    }
    s1[threadIdx.x] = s; s2[threadIdx.x] = ss;
    __syncthreads();
    for (int st = 128; st > 0; st >>= 1) {
        if (threadIdx.x < st) {
            s1[threadIdx.x] += s1[threadIdx.x + st];
            s2[threadIdx.x] += s2[threadIdx.x + st];
        }
        __syncthreads();
    }
    if (threadIdx.x == 0) {
        float m = s1[0] / (float)M;
        float v = s2[0] / (float)M - m * m;
        if (v < 0.0f) v = 0.0f;
        mean[o] = m;
        inv_std[o] = rsqrtf(v + EPSBN);
    }
}

// ---------------------------------------------------------------------------
// BN + ReLU, 2D grid (blockIdx.y = channel): no integer division, BN folded
// into a single FMA, float4 vectorized (M is always a multiple of 16).
// ---------------------------------------------------------------------------
__global__ void __launch_bounds__(256) bn_relu_kernel(float* __restrict__ Y,
                                                      const float* __restrict__ mean,
                                                      const float* __restrict__ inv,
                                                      const float* __restrict__ gamma,
                                                      const float* __restrict__ beta,
                                                      int M)
{
    int o = blockIdx.y;
    float sc = inv[o] * gamma[o];
    float sh = beta[o] - mean[o] * sc;
    float4* row = (float4*)(Y + (size_t)o * M);
    int n4 = M >> 2;
    int stride = gridDim.x * blockDim.x;
    for (int i = blockIdx.x * blockDim.x + threadIdx.x; i < n4; i += stride) {
        float4 v = row[i];
        v.x = fmaxf(v.x * sc + sh, 0.0f);
        v.y = fmaxf(v.y * sc + sh, 0.0f);
        v.z = fmaxf(v.z * sc + sh, 0.0f);
        v.w = fmaxf(v.w * sc + sh, 0.0f);
        row[i] = v;
    }
}

// out[o][g] = max_s Y[o][g*S + s]; 2D grid (blockIdx.y = o), float4 reads
// (S in {16,32,64,128}; rows are 16B aligned).
__global__ void __launch_bounds__(256) maxpool_kernel(const float* __restrict__ Y,
                                                      float* __restrict__ out,
                                                      int G, int S)
{
    int g = blockIdx.x * blockDim.x + threadIdx.x;
    int o = blockIdx.y;
    if (g >= G) return;
    const float4* p = (const float4*)(Y + ((size_t)o * G + g) * S);
    int n4 = S >> 2;
    float4 v = p[0];
    float m = fmaxf(fmaxf(v.x, v.y), fmaxf(v.z, v.w));
    for (int s = 1; s < n4; ++s) {
        float4 u = p[s];
        m = fmaxf(m, fmaxf(fmaxf(u.x, u.y), fmaxf(u.z, u.w)));
    }
    out[(size_t)o * G + g] = m;
}

// X3[(3+Cf)][B*Kp]: rows 0..2 = absolute coords2, rows 3.. = pooled feats
__global__ void build_x3_kernel(const float* __restrict__ coords2,
                                const float* __restrict__ F2,
                                float* __restrict__ X3,
                                int B, int Kp, int Cf)
{
    int M = B * Kp;
    int total = (3 + Cf) * M;
    int i = blockIdx.x * blockDim.x + threadIdx.x;
    if (i >= total) return;
    int c = i / M;
    int m = i % M;
    if (c < 3) {
        int b = m / Kp, k = m % Kp;
        X3[i] = coords2[((size_t)b * 3 + c) * Kp + k];
    } else {
        X3[i] = F2[(size_t)(c - 3) * M + m];
    }
}

// out[b][o] = Y[o][b]
__global__ void write_out_kernel(const float* __restrict__ Y,
                                 float* __restrict__ out,
                                 int O, int B)
{
    int i = blockIdx.x * blockDim.x + threadIdx.x;
    if (i >= O * B) return;
    int o = i % O, b = i / O;
    out[b * O + o] = Y[(size_t)o * B + b];
}

// ---------------------------------------------------------------------------
// Host orchestration
// ---------------------------------------------------------------------------
struct LayerP { const float *w, *b, *g, *be; int o, c; };

static void run_layer(const LayerP& L, const float* Xin, float* Yout, int M,
                      float* meanB, float* invB, hipStream_t st)
{
    dim3 grid(ceil_div(M, 16), ceil_div(L.o, 128));
    gemm_wmma_f32<<<grid, 256, 0, st>>>(L.w, Xin, L.b, Yout, L.o, L.c, M);
    stats_kernel<<<L.o, 256, 0, st>>>(Yout, M, meanB, invB);
    dim3 gbn(ceil_div(M >> 2, 256), L.o);
    bn_relu_kernel<<<gbn, 256, 0, st>>>(Yout, meanB, invB, L.g, L.be, M);
}

extern "C" void kernel_launch(void* const* d_in, const int* in_sizes, int n_in,
                              void* d_out, int out_size, void* d_ws, size_t ws_size,
                              hipStream_t stream)
{
    (void)in_sizes; (void)n_in; (void)out_size; (void)ws_size;
    const int B = 16;

    // ---- consume inputs in setup_inputs() insertion order ----
    int p = 0;
    const float* pt = (const float*)d_in[p++];                     // [16,3,4096]

    static const int chans1[3][3] = {{32, 32, 64}, {64, 64, 128}, {64, 96, 128}};
    static const int chans2[3][3] = {{64, 64, 128}, {128, 128, 256}, {128, 128, 256}};
    static const int chans3[3]    = {256, 512, 1024};
    static const float rad1[3] = {0.1f, 0.2f, 0.4f};
    static const int   ss1[3]  = {16, 32, 128};
    static const float rad2[3] = {0.2f, 0.4f, 0.8f};
    static const int   ss2[3]  = {32, 64, 128};

    LayerP msg1[3][3], msg2[3][3], msg3[3];
    for (int s = 0; s < 3; ++s) {
        int cin = 3;
        for (int l = 0; l < 3; ++l) {
            msg1[s][l].w = (const float*)d_in[p++];
            msg1[s][l].b = (const float*)d_in[p++];
            msg1[s][l].g = (const float*)d_in[p++];
            msg1[s][l].be = (const float*)d_in[p++];
            msg1[s][l].o = chans1[s][l]; msg1[s][l].c = cin; cin = chans1[s][l];
        }
    }
    for (int s = 0; s < 3; ++s) {
        int cin = 323;
        for (int l = 0; l < 3; ++l) {
            msg2[s][l].w = (const float*)d_in[p++];
            msg2[s][l].b = (const float*)d_in[p++];
            msg2[s][l].g = (const float*)d_in[p++];
            msg2[s][l].be = (const float*)d_in[p++];
            msg2[s][l].o = chans2[s][l]; msg2[s][l].c = cin; cin = chans2[s][l];
        }
    }
    {
        int cin = 643;
        for (int l = 0; l < 3; ++l) {
            msg3[l].w = (const float*)d_in[p++];
            msg3[l].b = (const float*)d_in[p++];
            msg3[l].g = (const float*)d_in[p++];
            msg3[l].be = (const float*)d_in[p++];
            msg3[l].o = chans3[l]; msg3[l].c = cin; cin = chans3[l];
        }
    }
    const float* fc1w = (const float*)d_in[p++]; const float* fc1b = (const float*)d_in[p++];
    const float* fc2w = (const float*)d_in[p++]; const float* fc2b = (const float*)d_in[p++];
    const float* fc3w = (const float*)d_in[p++]; const float* fc3b = (const float*)d_in[p++];
    const float* bn1g = (const float*)d_in[p++]; const float* bn1b = (const float*)d_in[p++];
    const float* bn2g = (const float*)d_in[p++]; const float* bn2b = (const float*)d_in[p++];

    // ---- carve workspace ----
    char* base = (char*)d_ws;
    size_t off = 0;
    auto carve = [&](size_t bytes) -> void* {
        void* q = (void*)(base + off);
        off += (bytes + 255) & ~(size_t)255;
        return q;
    };
    const size_t BIG = (size_t)128 * 1048576 * 4;   // 512 MB: worst layer 128ch x 1M
    float* buf0 = (float*)carve(BIG);
    float* buf1 = (float*)carve(BIG);
    float* pooled1 = (float*)carve((size_t)320 * 8192 * 4);   // [320][B*512]
    float* pooled2 = (float*)carve((size_t)640 * 2048 * 4);   // [640][B*128]
    float* pooled3 = (float*)carve((size_t)1024 * 16 * 4);    // [1024][B]
    float* fcY1 = (float*)carve((size_t)512 * 16 * 4);
    float* fcY2 = (float*)carve((size_t)256 * 16 * 4);
    float* fcY3 = (float*)carve((size_t)40 * 16 * 4);
    float* coords1 = (float*)carve((size_t)B * 3 * 512 * 4);
    float* coords2 = (float*)carve((size_t)B * 3 * 128 * 4);
    int* c1 = (int*)carve((size_t)B * 512 * 4);
    int* c2 = (int*)carve((size_t)B * 128 * 4);
    float* meanB = (float*)carve(1024 * 4);
    float* invB = (float*)carve(1024 * 4);

    // ---- stage 1: FPS on raw points, gather centroid coords ----
    fps_kernel<<<B, 256, 0, stream>>>(pt, c1, B, 4096, 512);
    gather_coords_kernel<<<ceil_div(B * 3 * 512, 256), 256, 0, stream>>>(pt, c1, coords1, B, 4096, 512);

    // ---- SA1 (MSG1): 3 scales over N=4096, K=512, Cf=0 ----
    {
        int K = 512, N = 4096;
        int coff = 0;
        for (int s = 0; s < 3; ++s) {
            int S = ss1[s];
            float r2 = rad1[s] * rad1[s];
            int M = B * K * S;
            group_kernel<<<ceil_div(B * K, 4), 128, 0, stream>>>(pt, nullptr, c1, buf0,
                                                                 B, N, K, S, 0, r2);
            const float* cur = buf0;
            float* dst = buf1;
            int Olast = 0;
            for (int l = 0; l < 3; ++l) {
                run_layer(msg1[s][l], cur, dst, M, meanB, invB, stream);
                cur = dst; Olast = msg1[s][l].o;
                dst = (dst == buf1) ? buf0 : buf1;
            }
            dim3 gp(ceil_div(B * K, 256), Olast);
            maxpool_kernel<<<gp, 256, 0, stream>>>(cur, pooled1 + (size_t)coff * (B * K), B * K, S);
            coff += Olast;
        }
    }

    // ---- stage 2: FPS on 512 centroids -> 128, gather coords ----
    fps_kernel<<<B, 256, 0, stream>>>(coords1, c2, B, 512, 128);
    gather_coords_kernel<<<ceil_div(B * 3 * 128, 256), 256, 0, stream>>>(coords1, c2, coords2, B, 512, 128);

    // ---- SA2 (MSG2): 3 scales over N=512, K=128, Cf=320 ----
    {
        int K = 128, N = 512, Cf = 320;
        int coff = 0;
        for (int s = 0; s < 3; ++s) {
            int S = ss2[s];
            float r2 = rad2[s] * rad2[s];
            int M = B * K * S;
            group_kernel<<<ceil_div(B * K, 4), 128, 0, stream>>>(coords1, pooled1, c2, buf0,
                                                                 B, N, K, S, Cf, r2);
            const float* cur = buf0;
            float* dst = buf1;
            int Olast = 0;
            for (int l = 0; l < 3; ++l) {
                run_layer(msg2[s][l], cur, dst, M, meanB, invB, stream);
                cur = dst; Olast = msg2[s][l].o;
                dst = (dst == buf1) ? buf0 : buf1;
            }
            dim3 gp(ceil_div(B * K, 256), Olast);
            maxpool_kernel<<<gp, 256, 0, stream>>>(cur, pooled2 + (size_t)coff * (B * K), B * K, S);
            coff += Olast;
        }
    }

    // ---- msg3: concat(coords2, pooled2) -> MLP [256,512,1024] -> max over 128 ----
    {
        int Kp = 128, M = B * Kp;
        build_x3_kernel<<<ceil_div(643 * M, 256), 256, 0, stream>>>(coords2, pooled2, buf0, B, Kp, 640);
        const float* cur = buf0;
        float* dst = buf1;
        int Olast = 0;
        for (int l = 0; l < 3; ++l) {
            run_layer(msg3[l], cur, dst, M, meanB, invB, stream);
            cur = dst; Olast = msg3[l].o;
            dst = (dst == buf1) ? buf0 : buf1;
        }
        dim3 gp(ceil_div(B, 256), Olast);
        maxpool_kernel<<<gp, 256, 0, stream>>>(cur, pooled3, B, Kp);
    }

    // ---- FC head (channel-major [O][B]) ----
    {
        LayerP L1 = {fc1w, fc1b, bn1g, bn1b, 512, 1024};
        run_layer(L1, pooled3, fcY1, B, meanB, invB, stream);
        LayerP L2 = {fc2w, fc2b, bn2g, bn2b, 256, 512};
        run_layer(L2, fcY1, fcY2, B, meanB, invB, stream);
        dim3 g3(ceil_div(B, 16), ceil_div(40, 128));
        gemm_wmma_f32<<<g3, 256, 0, stream>>>(fc3w, fcY2, fc3b, fcY3, 40, 256, B);
        write_out_kernel<<<ceil_div(40 * B, 256), 256, 0, stream>>>(fcY3, (float*)d_out, 40, B);
    }
}